// Deep6GCNModelAE_42855183679829
// MI455X (gfx1250) — compile-verified
//
#include <hip/hip_runtime.h>

typedef float v2f __attribute__((ext_vector_type(2)));
typedef float v8f __attribute__((ext_vector_type(8)));

#define NROWS 10000
#define MT    625      // NROWS / 16
#define JT    157      // ceil(625 / 4) j-strip groups for the decoder

// ---------------------------------------------------------------------------
// Generic fp32 WMMA GEMM:  C[M x Nw] = act(A[M x K]) @ W[K x Nw]
// One wave32 per 16x16 output tile; K stepped by 4 via V_WMMA_F32_16X16X4_F32.
// RELU_A fuses the ReLU that (in the reference dataflow) follows the previous
// spmm into the A-operand load of this GEMM.
// ---------------------------------------------------------------------------
template <bool RELU_A>
__global__ void gemm_wmma_f32(const float* __restrict__ A,
                              const float* __restrict__ W,
                              float* __restrict__ C,
                              int K, int Nw) {
  const int i0   = blockIdx.x * 16;
  const int n0   = blockIdx.y * 16;
  const int lane = threadIdx.x;   // 0..31, wave32
  const int lr   = lane & 15;
  const int hi   = lane >> 4;

  v8f c = {};
  for (int k = 0; k < K; k += 4) {
    const int ka = k + 2 * hi;
    // A fragment: lanes 0-15 hold row (i0+lr) K={k,k+1}; lanes 16-31 K={k+2,k+3}
    v2f a;
    a.x = A[(i0 + lr) * K + ka];
    a.y = A[(i0 + lr) * K + ka + 1];
    if (RELU_A) {
      a.x = fmaxf(a.x, 0.0f);
      a.y = fmaxf(a.y, 0.0f);
    }
    // B fragment: N = lr; lanes 0-15 hold K={k,k+1}; lanes 16-31 K={k+2,k+3}
    v2f b;
    b.x = W[ka * Nw + n0 + lr];
    b.y = W[(ka + 1) * Nw + n0 + lr];
    c = __builtin_amdgcn_wmma_f32_16x16x4_f32(false, a, false, b,
                                              (short)0, c, false, false);
  }
  // C/D layout: VGPR r -> M = r + 8*hi, N = lr
#pragma unroll
  for (int r = 0; r < 8; ++r) {
    C[(i0 + r + 8 * hi) * Nw + n0 + lr] = c[r];
  }
}

// ---------------------------------------------------------------------------
// COO SpMM scatter:  Y[row[e], f] += vals[e] * X[col[e], f]
// One thread per (edge, feature). F is a power of two (32 or 16).
// All feature matrices are ~1.3 MB -> L2 resident; atomics stay on-chip.
// ---------------------------------------------------------------------------
__global__ void spmm_scatter(const float* __restrict__ X,
                             const int* __restrict__ row,
                             const int* __restrict__ col,
                             const float* __restrict__ vals,
                             float* __restrict__ Y,
                             int E, int logF) {
  const long long t     = (long long)blockIdx.x * blockDim.x + threadIdx.x;
  const int       F     = 1 << logF;
  const long long total = (long long)E << logF;
  if (t >= total) return;
  const int e = (int)(t >> logF);
  const int f = (int)(t & (F - 1));
  atomicAdd(&Y[(long long)row[e] * F + f], vals[e] * X[(long long)col[e] * F + f]);
}

// ---------------------------------------------------------------------------
// Decoder: out = Z @ Z.T, Z is [NROWS x 16] fp32. Store-bandwidth bound
// (400 MB of fp32 -> ~17 us floor at 23.3 TB/s), so optimize the store path:
//  - one wave32 per 16x64 output strip (4 j-tiles, 16 WMMAs, A frags reused)
//  - C fragments staged through a 4 KB LDS tile, then written back as
//    float4 -> global_store_b128: each store instruction covers two
//    256 B contiguous segments (one per row; 40000 B row stride, so not all 256 B aligned).
// The partial last strip group uses a wave-uniform guard so EXEC remains
// all-ones around every WMMA (ISA requirement), with per-lane guards only
// on the final stores.
// ---------------------------------------------------------------------------
__global__ void decoder_zzT(const float* __restrict__ Z,
                            float* __restrict__ out) {
  __shared__ __align__(16) float tile[16 * 64];

  const int j0   = blockIdx.x * 64;   // strip of 4 j-tiles
  const int i0   = blockIdx.y * 16;
  const int lane = threadIdx.x;
  const int lr   = lane & 15;
  const int hi   = lane >> 4;

  // A fragments (Z rows i0..i0+15) for all 4 k-steps, loaded once.
  v2f a[4];
#pragma unroll
  for (int s = 0; s < 4; ++s) {
    const int kk = 4 * s + 2 * hi;
    a[s].x = Z[(i0 + lr) * 16 + kk];
    a[s].y = Z[(i0 + lr) * 16 + kk + 1];
  }

#pragma unroll
  for (int t = 0; t < 4; ++t) {
    const int jt = j0 + 16 * t;
    if (jt < NROWS) {               // uniform across the wave: EXEC stays all-1s
      v8f c = {};
#pragma unroll
      for (int s = 0; s < 4; ++s) {
        const int kk = 4 * s + 2 * hi;
        v2f b;                       // B[k, n] = Z[jt + n, k]
        b.x = Z[(jt + lr) * 16 + kk];
        b.y = Z[(jt + lr) * 16 + kk + 1];
        c = __builtin_amdgcn_wmma_f32_16x16x4_f32(false, a[s], false, b,
                                                  (short)0, c, false, false);
      }
#pragma unroll
      for (int r = 0; r < 8; ++r) {
        tile[(r + 8 * hi) * 64 + 16 * t + lr] = c[r];
      }
    }
  }
  __syncthreads();  // single-wave WG: S_NOP; DS ops are in-order within a wave

  // Coalesced writeback: 16 rows x 16 float4-chunks = 256 chunks, 8 iters.
  // Per instruction: lanes 0-15 -> row 2*it (256 B), lanes 16-31 -> row 2*it+1.
#pragma unroll
  for (int it = 0; it < 8; ++it) {
    const int idx = it * 32 + lane;
    const int r   = idx >> 4;        // row 0..15
    const int ch  = idx & 15;        // float4 chunk within row
    const int jc  = j0 + ch * 4;
    if (jc < NROWS) {
      const float4 v = *(const float4*)&tile[r * 64 + ch * 4];
      *(float4*)&out[(long long)(i0 + r) * NROWS + jc] = v;
    }
  }
}

// ---------------------------------------------------------------------------
extern "C" void kernel_launch(void* const* d_in, const int* in_sizes, int n_in,
                              void* d_out, int out_size, void* d_ws, size_t ws_size,
                              hipStream_t stream) {
  const float* features = (const float*)d_in[0];
  const int*   adj_row  = (const int*)d_in[1];
  const int*   adj_col  = (const int*)d_in[2];
  const float* adj_vals = (const float*)d_in[3];
  const float* Wl[7]    = {(const float*)d_in[4], (const float*)d_in[5],
                           (const float*)d_in[6], (const float*)d_in[7],
                           (const float*)d_in[8], (const float*)d_in[9],
                           (const float*)d_in[10]};
  float* out = (float*)d_out;
  const int E = in_sizes[1];

  float* XW = (float*)d_ws;          // GEMM output      (NROWS x 32 fp32)
  float* H  = XW + NROWS * 32;       // SpMM accumulator (NROWS x 32 fp32)

  const dim3 wave(32);
  const int  spmm32_blocks = (int)(((long long)E * 32 + 255) / 256);
  const int  spmm16_blocks = (int)(((long long)E * 16 + 255) / 256);

  // Layer 0: features(10000x512) @ W0(512x32) -> spmm -> (ReLU fused into next GEMM)
  gemm_wmma_f32<false><<<dim3(MT, 2), wave, 0, stream>>>(features, Wl[0], XW, 512, 32);
  hipMemsetAsync(H, 0, (size_t)NROWS * 32 * sizeof(float), stream);
  spmm_scatter<<<spmm32_blocks, 256, 0, stream>>>(XW, adj_row, adj_col, adj_vals, H, E, 5);

  // Layers 1..5: relu(H)(10000x32) @ Wi(32x32) -> spmm
  for (int l = 1; l <= 5; ++l) {
    gemm_wmma_f32<true><<<dim3(MT, 2), wave, 0, stream>>>(H, Wl[l], XW, 32, 32);
    hipMemsetAsync(H, 0, (size_t)NROWS * 32 * sizeof(float), stream);
    spmm_scatter<<<spmm32_blocks, 256, 0, stream>>>(XW, adj_row, adj_col, adj_vals, H, E, 5);
  }

  // Layer 6 (z_mean): relu(H) @ W6(32x16) -> spmm (no relu)
  gemm_wmma_f32<true><<<dim3(MT, 1), wave, 0, stream>>>(H, Wl[6], XW, 32, 16);
  hipMemsetAsync(H, 0, (size_t)NROWS * 16 * sizeof(float), stream);
  spmm_scatter<<<spmm16_blocks, 256, 0, stream>>>(XW, adj_row, adj_col, adj_vals, H, E, 4);

  // Decoder: out = Z @ Z.T  (store-bandwidth bound: 400 MB fp32)
  decoder_zzT<<<dim3(JT, MT), wave, 0, stream>>>(H, out);
}